// PhysicsMambaBlock_81862076662143
// MI455X (gfx1250) — compile-verified
//
#include <hip/hip_runtime.h>
#include <hip/hip_bf16.h>
#include <math.h>

// Problem constants (from the reference)
#define BATCH 16
#define SEQL  1024
#define DM    256
#define DS    64
#define DC    4
#define DI    512                     // EXP * DM
#define DTR   16                      // ceil(DM/16)
#define BL    (BATCH * SEQL)          // 16384 rows
#define NXZ   (2 * DI)                // 1024
#define NDBL  144                     // DTR + 2*DS (logical width)
#define NDBLP 192                     // padded x_dbl width: 3 waves * 64

#define KC    16                      // K chunk staged in LDS

typedef float v2f __attribute__((ext_vector_type(2)));
typedef float v8f __attribute__((ext_vector_type(8)));

// ---------------------------------------------------------------------------
// fp32 WMMA GEMM: C[M x N] = A[M x K] (row major) * B[K x N] (row major).
// N, K, WAVES are compile-time so the WMMA loop is completely branch-free:
// no divergent guards -> EXEC stays all-ones for every V_WMMA (ISA 7.12)
// and the 4 v8f accumulators never spill.
// Block = WAVES*32 threads -> 16 x (WAVES*64) output strip; each wave owns
// 16 x 64 (4 accumulators), so each A fragment feeds 4 WMMAs.
// Launch geometry must satisfy: gridDim.y * WAVES * 64 == N (exact cover).
// ---------------------------------------------------------------------------
template <int N, int K, int WAVES>
__global__ void wmma_gemm_f32(const float* __restrict__ A,
                              const float* __restrict__ B,
                              float* __restrict__ C) {
    constexpr int THREADS = WAVES * 32;
    constexpr int BNT     = WAVES * 64;        // block N tile

    __shared__ float Alds[16][KC + 1];         // +1 pad: conflict-free A frags
    __shared__ float Blds[KC][BNT];

    const int tid  = threadIdx.x;
    const int lane = tid & 31;
    const int wave = tid >> 5;
    const int row0 = blockIdx.x * 16;
    const int nb0  = blockIdx.y * BNT;
    const int wcol = wave * 64;                // wave's col offset in block tile
    const int half = lane >> 4;                // 0: lanes 0-15, 1: lanes 16-31
    const int q    = lane & 15;                // A row / B col / C col in tile

    v8f acc[4] = {};

    for (int k0 = 0; k0 < K; k0 += KC) {
        // ---- stage A (16 x KC), coalesced rows
        for (int i = tid; i < 16 * KC; i += THREADS)
            Alds[i >> 4][i & 15] = A[(size_t)(row0 + (i >> 4)) * K + k0 + (i & 15)];

        // ---- stage B (KC x BNT), coalesced, exactly 32 iters, no guards
#pragma unroll 4
        for (int i = tid; i < KC * BNT; i += THREADS) {
            const int r = i / BNT;
            const int c = i % BNT;
            Blds[r][c] = B[(size_t)(k0 + r) * N + nb0 + c];
        }

        // ---- prefetch hint for next B chunk (global_prefetch_b8)
        if (k0 + KC < K)
            __builtin_prefetch(&B[(size_t)(k0 + KC) * N + nb0 + (tid % BNT)], 0, 0);
        __syncthreads();

        // ---- 4 k-steps x 4 N-subtiles = 16 WMMAs per chunk, branch-free
#pragma unroll
        for (int kk = 0; kk < KC; kk += 4) {
            v2f a;
            a.x = Alds[q][kk + 2 * half + 0];
            a.y = Alds[q][kk + 2 * half + 1];
#pragma unroll
            for (int s = 0; s < 4; ++s) {
                v2f b;
                b.x = Blds[kk + 2 * half + 0][wcol + s * 16 + q];
                b.y = Blds[kk + 2 * half + 1][wcol + s * 16 + q];
                acc[s] = __builtin_amdgcn_wmma_f32_16x16x4_f32(
                    /*neg_a=*/false, a, /*neg_b=*/false, b,
                    /*c_mod=*/(short)0, acc[s],
                    /*reuse_a=*/false, /*reuse_b=*/false);
            }
        }
        __syncthreads();
    }

    // ---- store: VGPR r -> rows (r, r+8), col = q (ISA C/D layout)
#pragma unroll
    for (int s = 0; s < 4; ++s) {
        const int c0 = nb0 + wcol + s * 16;
#pragma unroll
        for (int r = 0; r < 8; ++r)
            C[(size_t)(row0 + r + 8 * half) * N + c0 + q] = acc[s][r];
    }
}

// ---------------------------------------------------------------------------
// Zero-pad w_x (512 x 144) into w_x_pad (512 x 192) so the x_dbl GEMM can
// run guard-free with N = 192 (cols 144..191 produce zeros, never read).
// ---------------------------------------------------------------------------
__global__ void pad_wx_kernel(const float* __restrict__ wx,
                              float* __restrict__ wxp) {
    const int i = blockIdx.x * blockDim.x + threadIdx.x;
    if (i >= DI * NDBLP) return;
    const int r = i / NDBLP;
    const int c = i % NDBLP;
    wxp[i] = (c < NDBL) ? wx[r * NDBL + c] : 0.0f;
}

// ---------------------------------------------------------------------------
// Depthwise causal conv (kernel DC=4, left pad 3) + bias + SiLU.
// Reads the first DI columns of xz, writes xc[BL x DI].
// ---------------------------------------------------------------------------
__global__ void conv_silu_kernel(const float* __restrict__ xz,
                                 const float* __restrict__ conv_w,
                                 const float* __restrict__ conv_b,
                                 float* __restrict__ xc) {
    const int idx = blockIdx.x * blockDim.x + threadIdx.x;
    if (idx >= BL * DI) return;
    const int d  = idx & (DI - 1);
    const int bl = idx >> 9;            // log2(DI) = 9
    const int l  = bl & (SEQL - 1);

    float acc = conv_b[d];
#pragma unroll
    for (int j = 0; j < DC; ++j) {
        const int lt = l + j - (DC - 1);
        if (lt >= 0)
            acc += xz[(size_t)(bl + j - (DC - 1)) * NXZ + d] * conv_w[d * DC + j];
    }
    const float s = acc / (1.0f + __expf(-acc));   // SiLU
    xc[(size_t)bl * DI + d] = s;
}

// ---------------------------------------------------------------------------
// Selective scan. Block = 128 threads, one (batch, channel) pair per thread.
// grid = (BATCH, DI/128). State h[64] and A-row a[64] live in VGPRs.
// Bm/Cm (shared across channels within a batch) are staged in LDS per step.
// Fuses: delta = softplus(dt*w_dt + b_dt), D skip, SiLU(z) gating.
// x_dbl has padded row stride NDBLP.
// ---------------------------------------------------------------------------
__global__ void scan_kernel(const float* __restrict__ xz,      // gate z at col DI+d
                            const float* __restrict__ xc,
                            const float* __restrict__ x_dbl,   // (BL, NDBLP)
                            const float* __restrict__ dt_vals, // (B, L, 1)
                            const float* __restrict__ w_dt,    // (1, DI)
                            const float* __restrict__ b_dt,    // (DI)
                            const float* __restrict__ A_log,   // (DI, DS)
                            const float* __restrict__ Dp,      // (DI)
                            float* __restrict__ y) {           // (BL, DI)
    __shared__ float ldsB[DS];
    __shared__ float ldsC[DS];

    const int b = blockIdx.x;
    const int d = blockIdx.y * blockDim.x + threadIdx.x;

    float h[DS];
    float a[DS];
#pragma unroll
    for (int n = 0; n < DS; ++n) {
        h[n] = 0.0f;
        a[n] = -__expf(A_log[(size_t)d * DS + n]);
    }
    const float wdt = w_dt[d];
    const float bdt = b_dt[d];
    const float Dv  = Dp[d];

    for (int l = 0; l < SEQL; ++l) {
        const int bl = b * SEQL + l;
        // stage Bm/Cm for this (b,l) into LDS (128 threads, 128 values)
        if (threadIdx.x < DS)
            ldsB[threadIdx.x] = x_dbl[(size_t)bl * NDBLP + DTR + threadIdx.x];
        else
            ldsC[threadIdx.x - DS] = x_dbl[(size_t)bl * NDBLP + DTR + DS + (threadIdx.x - DS)];
        __syncthreads();

        const float pre   = dt_vals[bl] * wdt + bdt;
        const float delta = (pre > 20.0f) ? pre : __logf(1.0f + __expf(pre));  // softplus
        const float xi    = xc[(size_t)bl * DI + d];
        const float dxi   = delta * xi;

        float yv = 0.0f;
#pragma unroll
        for (int n = 0; n < DS; ++n) {
            h[n] = __expf(delta * a[n]) * h[n] + dxi * ldsB[n];
            yv  += h[n] * ldsC[n];
        }

        const float zz   = xz[(size_t)bl * NXZ + DI + d];
        const float gate = zz / (1.0f + __expf(-zz));            // SiLU(z)
        y[(size_t)bl * DI + d] = (yv + xi * Dv) * gate;
        __syncthreads();
    }
}

// ---------------------------------------------------------------------------
// Launch
// ---------------------------------------------------------------------------
extern "C" void kernel_launch(void* const* d_in, const int* in_sizes, int n_in,
                              void* d_out, int out_size, void* d_ws, size_t ws_size,
                              hipStream_t stream) {
    const float* x       = (const float*)d_in[0];
    const float* dt_vals = (const float*)d_in[1];
    const float* w_in    = (const float*)d_in[2];
    const float* conv_w  = (const float*)d_in[3];
    const float* conv_b  = (const float*)d_in[4];
    const float* w_x     = (const float*)d_in[5];
    const float* w_dt    = (const float*)d_in[6];
    const float* b_dt    = (const float*)d_in[7];
    const float* A_log   = (const float*)d_in[8];
    const float* Dp      = (const float*)d_in[9];
    const float* w_out   = (const float*)d_in[10];
    float* out = (float*)d_out;

    // Workspace layout (bytes)
    char* ws = (char*)d_ws;
    const size_t SZ_XZ   = (size_t)BL * NXZ   * sizeof(float);  // 64 MB
    const size_t SZ_XC   = (size_t)BL * DI    * sizeof(float);  // 32 MB
    const size_t SZ_XDBL = (size_t)BL * NDBLP * sizeof(float);  // 12.6 MB
    const size_t SZ_WXP  = (size_t)DI * NDBLP * sizeof(float);  // 0.4 MB
    float* xz   = (float*)(ws);
    float* xc   = (float*)(ws + SZ_XZ);
    float* xdbl = (float*)(ws + SZ_XZ + SZ_XC);
    float* wxp  = (float*)(ws + SZ_XZ + SZ_XC + SZ_XDBL);
    float* yb   = (float*)(ws + SZ_XZ + SZ_XC + SZ_XDBL + SZ_WXP);  // 32 MB

    // 0) zero-pad w_x -> w_x_pad (512 x 192)
    pad_wx_kernel<<<(DI * NDBLP + 255) / 256, 256, 0, stream>>>(w_x, wxp);

    // 1) xz = x @ w_in          (16384x256 * 256x1024), 8 waves, grid.y=2
    wmma_gemm_f32<NXZ, DM, 8><<<dim3(BL / 16, NXZ / (8 * 64)), 256, 0, stream>>>(
        x, w_in, xz);

    // 2) causal depthwise conv + bias + SiLU -> xc
    conv_silu_kernel<<<(BL * DI) / 256, 256, 0, stream>>>(xz, conv_w, conv_b, xc);

    // 3) x_dbl = xc @ w_x_pad   (16384x512 * 512x192), 3 waves, grid.y=1
    wmma_gemm_f32<NDBLP, DI, 3><<<dim3(BL / 16, 1), 96, 0, stream>>>(
        xc, wxp, xdbl);

    // 4) selective scan + D skip + SiLU(z) gate -> yb
    scan_kernel<<<dim3(BATCH, DI / 128), dim3(128), 0, stream>>>(
        xz, xc, xdbl, dt_vals, w_dt, b_dt, A_log, Dp, yb);

    // 5) out = yb @ w_out       (16384x512 * 512x256), 4 waves, grid.y=1
    wmma_gemm_f32<DM, DI, 4><<<dim3(BL / 16, 1), 128, 0, stream>>>(
        yb, w_out, out);
}